// PixelflyLinear_32203664785504
// MI455X (gfx1250) — compile-verified
//
#include <hip/hip_runtime.h>

typedef __attribute__((ext_vector_type(2))) float v2f;
typedef __attribute__((ext_vector_type(8))) float v8f;

#define TOKENS  4096
#define IN_F    4096
#define OUT_F   4096
#define LOWRANK 256
#define BS      256
#define ACTIVE  5
#define KBF     (ACTIVE * BS)   // 1280

// ---------------------------------------------------------------------------
// Kernel 1: gather butterfly weights into Wg[f][j*256+c] (row-major [4096,1280])
// Wg[ob*256+rr][j*256+c] = weight[q*256 + k/5][k%5][c],  k = (flat%5)*256+rr,
// q = flat/5, flat = flat_idx[ob*5+j].
// ---------------------------------------------------------------------------
__global__ __launch_bounds__(256) void gather_wg(const float* __restrict__ weight,
                                                 const int* __restrict__ flat,
                                                 float* __restrict__ wg) {
    int g   = blockIdx.x * 256 + threadIdx.x;   // [0, OUT_F*KBF)
    int f   = g / KBF;
    int rem = g - f * KBF;
    int j   = rem >> 8;                         // chunk 0..4
    int c   = rem & 255;
    int ob  = f >> 8;
    int rr  = f & 255;
    int fl  = flat[ob * ACTIVE + j];
    int q   = fl / ACTIVE;
    int bb  = fl - q * ACTIVE;
    int k   = bb * BS + rr;
    int kr  = k / ACTIVE;
    int a   = k - kr * ACTIVE;
    int srcRow = q * BS + kr;
    wg[g] = weight[(srcRow * ACTIVE + a) * BS + c];
}

// ---------------------------------------------------------------------------
// Kernel 2: T[4096,256] = x @ w1.T    (wave tile: 32 tokens x 32 features)
// 4 independent accumulators -> WMMA->WMMA accumulator RAW distance of 4,
// 4 float2 loads per 4 WMMAs, A/B fragments each reused twice.
// A frag: lane l -> float2 x [t0 + 16i + (l&15)][k + 2*(l>>4)]
// B frag: lane l -> float2 w1[f0 + 16n + (l&15)][k + 2*(l>>4)]
// ---------------------------------------------------------------------------
__global__ __launch_bounds__(256) void gemm_xw1(const float* __restrict__ x,
                                                const float* __restrict__ w1,
                                                float* __restrict__ T) {
    const int lane = threadIdx.x & 31;
    const int wave = threadIdx.x >> 5;
    const int tile = blockIdx.x * 8 + wave;     // 1024 tiles total
    const int nft  = LOWRANK / 32;              // 8
    const int tt   = tile / nft;                // token tile (32 tokens)
    const int ft   = tile - tt * nft;           // feature tile (32 features)
    const int t0   = tt * 32;
    const int f0   = ft * 32;
    const int lr   = lane & 15;
    const int co   = (lane >> 4) * 2;

    const float* a0p = x  + (t0 + lr)      * IN_F + co;
    const float* a1p = x  + (t0 + 16 + lr) * IN_F + co;
    const float* b0p = w1 + (f0 + lr)      * IN_F + co;
    const float* b1p = w1 + (f0 + 16 + lr) * IN_F + co;

    v8f acc00 = {}, acc01 = {}, acc10 = {}, acc11 = {};
    for (int k = 0; k < IN_F; k += 16) {
#pragma unroll
        for (int u = 0; u < 16; u += 4) {
            v2f a0 = *(const v2f*)(a0p + k + u);
            v2f a1 = *(const v2f*)(a1p + k + u);
            v2f b0 = *(const v2f*)(b0p + k + u);
            v2f b1 = *(const v2f*)(b1p + k + u);
            acc00 = __builtin_amdgcn_wmma_f32_16x16x4_f32(false, a0, false, b0,
                                                          (short)0, acc00, false, false);
            acc01 = __builtin_amdgcn_wmma_f32_16x16x4_f32(false, a0, false, b1,
                                                          (short)0, acc01, false, false);
            acc10 = __builtin_amdgcn_wmma_f32_16x16x4_f32(false, a1, false, b0,
                                                          (short)0, acc10, false, false);
            acc11 = __builtin_amdgcn_wmma_f32_16x16x4_f32(false, a1, false, b1,
                                                          (short)0, acc11, false, false);
        }
    }
    // D layout: m = r + 8*(lane>>4), n = lane&15
    const int m0 = (lane >> 4) * 8;
#pragma unroll
    for (int r = 0; r < 8; ++r) {
        T[(t0 + m0 + r)      * LOWRANK + f0 + lr]      = acc00[r];
        T[(t0 + m0 + r)      * LOWRANK + f0 + 16 + lr] = acc01[r];
        T[(t0 + 16 + m0 + r) * LOWRANK + f0 + lr]      = acc10[r];
        T[(t0 + 16 + m0 + r) * LOWRANK + f0 + 16 + lr] = acc11[r];
    }
}

// ---------------------------------------------------------------------------
// Kernel 3: fused output. Wave tile 64 tokens x 64 features.
// out = butterfly(x, Wg) + T @ w2.T + b
// 16 independent v8f accumulators; per K-step of 4: 8 float2 loads, 16 WMMAs.
// ---------------------------------------------------------------------------
__global__ __launch_bounds__(256) void fused_out(const float* __restrict__ x,
                                                 const float* __restrict__ wg,
                                                 const float* __restrict__ T,
                                                 const float* __restrict__ w2,
                                                 const float* __restrict__ bias,
                                                 const int* __restrict__ flat,
                                                 float* __restrict__ out) {
    const int lane = threadIdx.x & 31;
    const int wave = threadIdx.x >> 5;
    const int tile = blockIdx.x * 8 + wave;     // 4096 tiles total
    const int nft  = OUT_F / 64;                // 64
    const int tt   = tile / nft;
    const int ft   = tile - tt * nft;
    const int t0   = tt * 64;
    const int f0   = ft * 64;
    const int ob   = f0 >> 8;                   // 64 | 256 -> fixed per tile
    const int lr   = lane & 15;
    const int co   = (lane >> 4) * 2;

    v8f acc[4][4];
#pragma unroll
    for (int i = 0; i < 4; ++i)
#pragma unroll
        for (int n = 0; n < 4; ++n) acc[i][n] = (v8f){};

    // ---- butterfly: 5 K-chunks of 256, x columns chosen by q_j ----
    for (int j = 0; j < ACTIVE; ++j) {
        const int q = flat[ob * ACTIVE + j] / ACTIVE;   // wave-uniform
        const float* arow[4];
        const float* brow[4];
#pragma unroll
        for (int i = 0; i < 4; ++i)
            arow[i] = x + (t0 + 16 * i + lr) * IN_F + q * BS + co;
#pragma unroll
        for (int n = 0; n < 4; ++n)
            brow[n] = wg + (f0 + 16 * n + lr) * KBF + j * BS + co;

        for (int k = 0; k < BS; k += 4) {
            v2f af[4], bf[4];
#pragma unroll
            for (int i = 0; i < 4; ++i) af[i] = *(const v2f*)(arow[i] + k);
#pragma unroll
            for (int n = 0; n < 4; ++n) bf[n] = *(const v2f*)(brow[n] + k);
#pragma unroll
            for (int i = 0; i < 4; ++i)
#pragma unroll
                for (int n = 0; n < 4; ++n)
                    acc[i][n] = __builtin_amdgcn_wmma_f32_16x16x4_f32(
                        false, af[i], false, bf[n], (short)0, acc[i][n], false, false);
        }
    }

    // ---- lowrank: K = 256 chunk from T and w2 ----
    {
        const float* arow[4];
        const float* brow[4];
#pragma unroll
        for (int i = 0; i < 4; ++i)
            arow[i] = T  + (t0 + 16 * i + lr) * LOWRANK + co;
#pragma unroll
        for (int n = 0; n < 4; ++n)
            brow[n] = w2 + (f0 + 16 * n + lr) * LOWRANK + co;

        for (int k = 0; k < LOWRANK; k += 4) {
            v2f af[4], bf[4];
#pragma unroll
            for (int i = 0; i < 4; ++i) af[i] = *(const v2f*)(arow[i] + k);
#pragma unroll
            for (int n = 0; n < 4; ++n) bf[n] = *(const v2f*)(brow[n] + k);
#pragma unroll
            for (int i = 0; i < 4; ++i)
#pragma unroll
                for (int n = 0; n < 4; ++n)
                    acc[i][n] = __builtin_amdgcn_wmma_f32_16x16x4_f32(
                        false, af[i], false, bf[n], (short)0, acc[i][n], false, false);
        }
    }

    // ---- bias + store ----
    const int m0 = (lane >> 4) * 8;
#pragma unroll
    for (int n = 0; n < 4; ++n) {
        const float bv = bias[f0 + 16 * n + lr];
#pragma unroll
        for (int i = 0; i < 4; ++i)
#pragma unroll
            for (int r = 0; r < 8; ++r)
                out[(t0 + 16 * i + m0 + r) * OUT_F + f0 + 16 * n + lr] = acc[i][n][r] + bv;
    }
}

// ---------------------------------------------------------------------------
// Launch. Workspace layout (floats): Wg [4096*1280] (20 MB), T [4096*256] (4 MB)
// ---------------------------------------------------------------------------
extern "C" void kernel_launch(void* const* d_in, const int* in_sizes, int n_in,
                              void* d_out, int out_size, void* d_ws, size_t ws_size,
                              hipStream_t stream) {
    const float* x      = (const float*)d_in[0];
    const float* weight = (const float*)d_in[1];
    const float* w1     = (const float*)d_in[2];
    const float* w2     = (const float*)d_in[3];
    const float* bias   = (const float*)d_in[4];
    const int*   flat   = (const int*)d_in[5];
    float*       out    = (float*)d_out;

    float* wg = (float*)d_ws;                       // 4096*1280 floats
    float* T  = wg + (size_t)OUT_F * KBF;           // 4096*256 floats

    gather_wg<<<dim3((OUT_F * KBF) / 256), dim3(256), 0, stream>>>(weight, flat, wg);

    // 1024 tiles (128 token-tiles x 8 feature-tiles), 8 waves/block
    gemm_xw1<<<dim3(1024 / 8), dim3(256), 0, stream>>>(x, w1, T);

    // 4096 tiles (64 x 64), 8 waves/block
    fused_out<<<dim3(4096 / 8), dim3(256), 0, stream>>>(x, wg, T, w2, bias, flat, out);
}